// GraphConvolution_14061722927710
// MI455X (gfx1250) — compile-verified
//
#include <hip/hip_runtime.h>
#include <stdint.h>

typedef __attribute__((ext_vector_type(2))) float v2f;
typedef __attribute__((ext_vector_type(8))) float v8f;

#define D 128           // D_IN == D_OUT == 128
#define TILE_M 16

// ---------------------------------------------------------------------------
// Kernel 1: h = x @ W using V_WMMA_F32_16X16X4_F32.
// Grid: one block per 16-row strip of nodes (3125 blocks), 256 threads = 8
// waves; wave w computes the 16x16 tile at columns [16w, 16w+16).
// x strip (16x128 f32 = 8KB) staged in LDS and shared by all 8 waves.
// ---------------------------------------------------------------------------
__global__ void __launch_bounds__(256)
gemm_h_kernel(const float* __restrict__ x, const float* __restrict__ w,
              float* __restrict__ h, int n_nodes) {
  __shared__ float lds_a[TILE_M * D];   // 8 KB

  const int row0 = blockIdx.x * TILE_M;
  const int tid  = threadIdx.x;

  // Cooperative load of the 16x128 x-strip: 512 float4, 2 per thread.
  {
    const float4* src = (const float4*)(x + (size_t)row0 * D);
    float4* dst = (float4*)lds_a;
    dst[tid]       = src[tid];
    dst[tid + 256] = src[tid + 256];
  }
  __syncthreads();

  const int wave = tid >> 5;          // 0..7 -> N tile
  const int lane = tid & 31;
  const int m    = lane & 15;         // A row
  const int kk   = (lane >> 4) << 1;  // lanes 0-15: K={0,1}; 16-31: K={2,3}
  const int n    = (wave << 4) + (lane & 15);  // B/D column

  v8f c = {};
  #pragma unroll
  for (int k0 = 0; k0 < D; k0 += 4) {
    v2f a, b;
    a.x = lds_a[m * D + k0 + kk];
    a.y = lds_a[m * D + k0 + kk + 1];
    b.x = w[(size_t)(k0 + kk) * D + n];
    b.y = w[(size_t)(k0 + kk + 1) * D + n];
    // D = A(16x4) * B(4x16) + C ; 8-arg form:
    // (neg_a, A, neg_b, B, c_mod, C, reuse_a, reuse_b)
    c = __builtin_amdgcn_wmma_f32_16x16x4_f32(false, a, false, b,
                                              (short)0, c, false, false);
  }

  // C/D layout: VGPR v holds M=v (lanes 0-15) / M=v+8 (lanes 16-31), N=lane&15.
  const int mbase = (lane >> 4) << 3;
  float* out = h + (size_t)(row0 + mbase) * D + n;
  #pragma unroll
  for (int v = 0; v < 8; ++v) out[(size_t)v * D] = c[v];
}

// ---------------------------------------------------------------------------
// Kernel 2: out[n][d] = bias[d]  (d_out is poisoned before timing)
// ---------------------------------------------------------------------------
__global__ void __launch_bounds__(256)
init_bias_kernel(float* __restrict__ out, const float* __restrict__ bias,
                 int n_nodes) {
  size_t i = (size_t)blockIdx.x * blockDim.x + threadIdx.x;   // float4 index
  size_t total = (size_t)n_nodes * (D / 4);
  if (i >= total) return;
  const float4* b4 = (const float4*)bias;
  ((float4*)out)[i] = b4[i & (D / 4 - 1)];
}

// ---------------------------------------------------------------------------
// Kernel 3: one wave per edge. Gather h[col] (float4/lane, 128 floats/wave),
// scatter-add into out[row] with non-returning global f32 atomics.
// ---------------------------------------------------------------------------
__global__ void __launch_bounds__(256)
edge_scatter_kernel(const float* __restrict__ h,
                    const int64_t* __restrict__ edge_index,
                    float* __restrict__ out, int n_edges) {
  int e = blockIdx.x * (blockDim.x >> 5) + (threadIdx.x >> 5);
  if (e >= n_edges) return;
  const int lane = threadIdx.x & 31;

  const int64_t dst = edge_index[e];             // row: scatter target
  const int64_t src = edge_index[n_edges + e];   // col: gather source

  float4 v = ((const float4*)(h + (size_t)src * D))[lane];
  float* o = out + (size_t)dst * D + lane * 4;
  atomicAdd(o + 0, v.x);
  atomicAdd(o + 1, v.y);
  atomicAdd(o + 2, v.z);
  atomicAdd(o + 3, v.w);
}

// ---------------------------------------------------------------------------
extern "C" void kernel_launch(void* const* d_in, const int* in_sizes, int n_in,
                              void* d_out, int out_size, void* d_ws, size_t ws_size,
                              hipStream_t stream) {
  const float*   x    = (const float*)d_in[0];     // [N, 128]
  const int64_t* ei   = (const int64_t*)d_in[1];   // [2, E] int64
  const float*   w    = (const float*)d_in[2];     // [128, 128]
  const float*   bias = (const float*)d_in[3];     // [128]
  float* out = (float*)d_out;                      // [N, 128]
  float* h   = (float*)d_ws;                       // [N, 128] scratch (25.6 MB)

  const int n_nodes = in_sizes[0] / D;
  const int n_edges = in_sizes[1] / 2;

  // 1) h = x @ W   (one 16-row strip per block, 8 waves cover 8 N-tiles)
  {
    dim3 grid((n_nodes + TILE_M - 1) / TILE_M);
    gemm_h_kernel<<<grid, 256, 0, stream>>>(x, w, h, n_nodes);
  }

  // 2) out = broadcast(bias)
  {
    size_t total = (size_t)n_nodes * (D / 4);
    dim3 grid((unsigned)((total + 255) / 256));
    init_bias_kernel<<<grid, 256, 0, stream>>>(out, bias, n_nodes);
  }

  // 3) out[row] += h[col] over all edges (wave-per-edge)
  {
    const int waves_per_block = 256 / 32;
    dim3 grid((n_edges + waves_per_block - 1) / waves_per_block);
    edge_scatter_kernel<<<grid, 256, 0, stream>>>(h, ei, out, n_edges);
  }
}